// InvariantMessagePasser_1245540515970
// MI455X (gfx1250) — compile-verified
//
#include <hip/hip_runtime.h>

typedef __attribute__((ext_vector_type(2))) float v2f;
typedef __attribute__((ext_vector_type(8))) float v8f;

#define CHUNK 256  // edges staged in LDS per pass (int2 -> 2 KB)

// ---------------- utility kernels ----------------

__global__ void zero_f32_kernel(float* __restrict__ p, long n) {
  long i = (long)blockIdx.x * blockDim.x + threadIdx.x;
  long stride = (long)gridDim.x * blockDim.x;
  for (; i < n; i += stride) p[i] = 0.0f;
}

__global__ void zero_i32_kernel(int* __restrict__ p, int n) {
  int i = blockIdx.x * blockDim.x + threadIdx.x;
  if (i < n) p[i] = 0;
}

__global__ void count_kernel(const int* __restrict__ centers,
                             int* __restrict__ counts, int E) {
  int i = blockIdx.x * blockDim.x + threadIdx.x;
  if (i < E) atomicAdd(&counts[centers[i]], 1);
}

// Single-block exclusive scan of counts -> offsets; resets counts to 0 for
// reuse as the scatter cursor.
__global__ __launch_bounds__(1024) void scan_kernel(int* __restrict__ counts,
                                                    int* __restrict__ offsets,
                                                    int nAtoms) {
  __shared__ int buf[1024];
  __shared__ int carryS;
  if (threadIdx.x == 0) carryS = 0;
  __syncthreads();
  for (int base = 0; base < nAtoms; base += 1024) {
    int i = base + threadIdx.x;
    int v = (i < nAtoms) ? counts[i] : 0;
    buf[threadIdx.x] = v;
    __syncthreads();
    for (int off = 1; off < 1024; off <<= 1) {
      int t = (threadIdx.x >= off) ? buf[threadIdx.x - off] : 0;
      __syncthreads();
      buf[threadIdx.x] += t;
      __syncthreads();
    }
    int carry = carryS;
    int excl = buf[threadIdx.x] - v + carry;
    if (i < nAtoms) { offsets[i] = excl; counts[i] = 0; }
    __syncthreads();  // everyone done reading carryS / buf
    if (threadIdx.x == 1023) carryS = carry + buf[1023];
    __syncthreads();
  }
  if (threadIdx.x == 0) offsets[nAtoms] = carryS;
}

__global__ void scatter_kernel(const int* __restrict__ centers,
                               const int* __restrict__ offsets,
                               int* __restrict__ cursor,
                               int* __restrict__ perm, int E) {
  int i = blockIdx.x * blockDim.x + threadIdx.x;
  if (i < E) {
    int c = centers[i];
    int pos = offsets[c] + atomicAdd(&cursor[c], 1);
    perm[pos] = i;
  }
}

// ---------------- WMMA accumulation kernel ----------------
// One workgroup (8 waves) per atom.  Edges owned by the atom are staged into
// LDS in chunks of 256 (edge id + neighbor id, pad slots clamped to the last
// valid edge).  Each wave owns up to 8 of the 40 16-channel output tiles and
// contracts the staged edges 4 at a time with V_WMMA_F32_16X16X4_F32:
//   out[a][m, ch] = sum_e sh_l[e, m] * (rb_l[e, ch] * emb[nbr(e), ch])
// All loads in the hot loop are unconditional (32-bit indexed); pad edges are
// masked only on the B operand (A x 0 = 0, and garbage A rows >= 2l+1 fall in
// C rows that are never stored), so EXEC stays all-ones around every WMMA.
__global__ __launch_bounds__(256) void wmma_msg_kernel(
    const float* __restrict__ sh0, const float* __restrict__ sh1,
    const float* __restrict__ sh2, const float* __restrict__ sh3,
    const float* __restrict__ rb0, const float* __restrict__ rb1,
    const float* __restrict__ rb2, const float* __restrict__ rb3,
    const float* __restrict__ emb, const int* __restrict__ neighbors,
    const int* __restrict__ offsets, const int* __restrict__ perm,
    float* __restrict__ out, int kmax,
    int t1, int t2, int t3, int T,
    int k0, int k1, int k2, int k3,
    int r0, int r1, int r2, int r3,
    int o1, int o2, int o3) {
  __shared__ int2 sEN[CHUNK];

  int a = blockIdx.x;
  int start = offsets[a];
  int n = offsets[a + 1] - start;  // edges owned by this atom
  if (n <= 0) return;              // uniform over the whole block

  const int nw = 8;  // blockDim.x / 32
  int wave = threadIdx.x >> 5;
  int lane = threadIdx.x & 31;
  int half = lane >> 4;  // K-slot pair selector
  int lm = lane & 15;    // M (A rows) / N (B,C cols)

  v8f accs[8];
#pragma unroll
  for (int tIdx = 0; tIdx < 8; ++tIdx) accs[tIdx] = (v8f){};

  for (int cb = 0; cb < n; cb += CHUNK) {
    int cn = n - cb;
    if (cn > CHUNK) cn = CHUNK;
    int cnPad = (cn + 3) & ~3;

    // cooperative staging: clamp pad slots to the last valid edge
    for (int i = threadIdx.x; i < cnPad; i += blockDim.x) {
      int src = cb + (i < cn ? i : cn - 1);
      int e = perm[start + src];
      sEN[i] = make_int2(e, neighbors[e]);
      __builtin_prefetch(&perm[start + src + CHUNK], 0, 1);  // speculative
    }
    __syncthreads();

#pragma unroll
    for (int tIdx = 0; tIdx < 8; ++tIdx) {
      int t = wave + tIdx * nw;  // wave-uniform
      if (t < T) {
        const float* sh; const float* rb; int rows, kl, ch0, ob;
        if (t < t1)      { sh=sh0; rb=rb0; rows=r0; kl=k0; ch0=t*16;        ob=0;  }
        else if (t < t2) { sh=sh1; rb=rb1; rows=r1; kl=k1; ch0=(t-t1)*16; ob=o1; }
        else if (t < t3) { sh=sh2; rb=rb2; rows=r2; kl=k2; ch0=(t-t2)*16; ob=o2; }
        else             { sh=sh3; rb=rb3; rows=r3; kl=k3; ch0=(t-t3)*16; ob=o3; }

        int ch = ch0 + lm;
        int rowc = (lm < rows) ? lm : rows - 1;  // clamped A-row (garbage rows unstored)
        v8f acc = accs[tIdx];
        for (int j = 0; j < cnPad; j += 4) {
          int iA = j + (half << 1);  // this lane's two K slots: iA, iA+1
          int2 en0 = sEN[iA];
          int2 en1 = sEN[iA + 1];
          float ax = sh[en0.x * rows + rowc];
          float ay = sh[en1.x * rows + rowc];
          float w0 = rb[en0.x * kl + ch] * emb[en0.y * kmax + ch];
          float w1 = rb[en1.x * kl + ch] * emb[en1.y * kmax + ch];
          float bx = (iA < cn) ? w0 : 0.f;      // v_cndmask, no branch
          float by = (iA + 1 < cn) ? w1 : 0.f;  // v_cndmask, no branch
          v2f A = {ax, ay};
          v2f B = {bx, by};
          acc = __builtin_amdgcn_wmma_f32_16x16x4_f32(
              false, A, false, B, (short)0, acc, false, false);
        }
        accs[tIdx] = acc;
      }
    }
    __syncthreads();
  }

  // C layout: lanes 0-15 hold M = vgpr index (0..7), N = lane.  rows <= 7 so
  // only the low half-wave carries valid rows; exclusive ownership -> stores.
#pragma unroll
  for (int tIdx = 0; tIdx < 8; ++tIdx) {
    int t = wave + tIdx * nw;
    if (t < T && half == 0) {
      int rows, kl, ch0, ob;
      if (t < t1)      { rows=r0; kl=k0; ch0=t*16;        ob=0;  }
      else if (t < t2) { rows=r1; kl=k1; ch0=(t-t1)*16; ob=o1; }
      else if (t < t3) { rows=r2; kl=k2; ch0=(t-t2)*16; ob=o2; }
      else             { rows=r3; kl=k3; ch0=(t-t3)*16; ob=o3; }
      float* op = out + ob + a * rows * kl + ch0 + lm;
#pragma unroll
      for (int v = 0; v < 8; ++v)
        if (v < rows) op[v * kl] = accs[tIdx][v];
    }
  }
}

// ---------------- fallback: atomic scatter (if workspace too small) --------

__device__ __forceinline__ void edge_part(const float* __restrict__ sh,
                                          const float* __restrict__ rb,
                                          const float* __restrict__ emb,
                                          float* __restrict__ outl,
                                          int e, int c, int nb,
                                          int rows, int kl, int kmax) {
  int nelem = rows * kl;
  for (int idx = threadIdx.x; idx < nelem; idx += blockDim.x) {
    int m = idx / kl;
    int ch = idx - m * kl;
    float v = sh[e * rows + m] * rb[e * kl + ch] * emb[nb * kmax + ch];
    atomicAdd(&outl[c * nelem + idx], v);
  }
}

__global__ __launch_bounds__(256) void edge_atomic_kernel(
    const float* __restrict__ sh0, const float* __restrict__ sh1,
    const float* __restrict__ sh2, const float* __restrict__ sh3,
    const float* __restrict__ rb0, const float* __restrict__ rb1,
    const float* __restrict__ rb2, const float* __restrict__ rb3,
    const float* __restrict__ emb, const int* __restrict__ centers,
    const int* __restrict__ neighbors, float* __restrict__ out,
    int E, int kmax,
    int k0, int k1, int k2, int k3,
    int r0, int r1, int r2, int r3,
    int o1, int o2, int o3) {
  int e = blockIdx.x;
  if (e >= E) return;
  int c = centers[e];
  int nb = neighbors[e];
  edge_part(sh0, rb0, emb, out,      e, c, nb, r0, k0, kmax);
  edge_part(sh1, rb1, emb, out + o1, e, c, nb, r1, k1, kmax);
  edge_part(sh2, rb2, emb, out + o2, e, c, nb, r2, k2, kmax);
  edge_part(sh3, rb3, emb, out + o3, e, c, nb, r3, k3, kmax);
}

// ---------------- launcher ----------------

extern "C" void kernel_launch(void* const* d_in, const int* in_sizes, int n_in,
                              void* d_out, int out_size, void* d_ws,
                              size_t ws_size, hipStream_t stream) {
  // setup_inputs() dict order: sh_l0, rb_l0, sh_l1, rb_l1, sh_l2, rb_l2,
  // sh_l3, rb_l3, embedding, centers, neighbors, n_atoms
  const float* sh0 = (const float*)d_in[0];
  const float* rb0 = (const float*)d_in[1];
  const float* sh1 = (const float*)d_in[2];
  const float* rb1 = (const float*)d_in[3];
  const float* sh2 = (const float*)d_in[4];
  const float* rb2 = (const float*)d_in[5];
  const float* sh3 = (const float*)d_in[6];
  const float* rb3 = (const float*)d_in[7];
  const float* emb = (const float*)d_in[8];
  const int* centers = (const int*)d_in[9];
  const int* neighbors = (const int*)d_in[10];
  float* out = (float*)d_out;

  int E = in_sizes[9];
  int r0 = in_sizes[0] / E, r1 = in_sizes[2] / E;
  int r2 = in_sizes[4] / E, r3 = in_sizes[6] / E;
  int k0 = in_sizes[1] / E, k1 = in_sizes[3] / E;
  int k2 = in_sizes[5] / E, k3 = in_sizes[7] / E;
  int kmax = k0;
  int nAtoms = in_sizes[8] / kmax;
  int o1 = nAtoms * r0 * k0;
  int o2 = o1 + nAtoms * r1 * k1;
  int o3 = o2 + nAtoms * r2 * k2;

  // zero the (poisoned) output every call
  {
    long n = (long)out_size;
    long b = (n + 255) / 256;
    int blocks = (int)(b > 16384 ? 16384 : b);
    zero_f32_kernel<<<blocks, 256, 0, stream>>>(out, n);
  }

  size_t need = sizeof(int) * ((size_t)(nAtoms + 1) + (size_t)nAtoms + (size_t)E);
  int t1 = k0 / 16;
  int t2 = t1 + k1 / 16;
  int t3 = t2 + k2 / 16;
  int T = t3 + k3 / 16;

  if (ws_size >= need && T <= 64) {
    int* offsets = (int*)d_ws;             // nAtoms + 1
    int* cursor = offsets + (nAtoms + 1);  // nAtoms (counts, then cursor)
    int* perm = cursor + nAtoms;           // E

    zero_i32_kernel<<<(nAtoms + 255) / 256, 256, 0, stream>>>(cursor, nAtoms);
    count_kernel<<<(E + 255) / 256, 256, 0, stream>>>(centers, cursor, E);
    scan_kernel<<<1, 1024, 0, stream>>>(cursor, offsets, nAtoms);
    scatter_kernel<<<(E + 255) / 256, 256, 0, stream>>>(centers, offsets,
                                                        cursor, perm, E);
    wmma_msg_kernel<<<nAtoms, 256, 0, stream>>>(
        sh0, sh1, sh2, sh3, rb0, rb1, rb2, rb3, emb, neighbors, offsets, perm,
        out, kmax, t1, t2, t3, T, k0, k1, k2, k3, r0, r1, r2, r3, o1, o2, o3);
  } else {
    edge_atomic_kernel<<<E, 256, 0, stream>>>(
        sh0, sh1, sh2, sh3, rb0, rb1, rb2, rb3, emb, centers, neighbors, out,
        E, kmax, k0, k1, k2, k3, r0, r1, r2, r3, o1, o2, o3);
  }
}